// APPNPNet_80676665688551
// MI455X (gfx1250) — compile-verified
//
#include <hip/hip_runtime.h>

typedef __attribute__((ext_vector_type(2))) float v2f;
typedef __attribute__((ext_vector_type(8))) float v8f;

#define FIN   256
#define HID   256
#define FOUT  64
#define ALPHA 0.1f
#define BETA  0.9f
#define HSTR  260   // LDS row stride (floats), pad to spread across 64 banks

#define WMMA_F32(a, b, c) \
    __builtin_amdgcn_wmma_f32_16x16x4_f32(false, (a), false, (b), (short)0, (c), false, false)

// ---------------------------------------------------------------------------
// Fused MLP: z = (relu(x @ w1^T + b1)) @ w2^T + b2
// One wave32 owns a 16-row strip. f32 WMMA 16x16x4 keeps reference precision.
// Register-blocked 4 N-tiles wide: each A fragment is loaded once per k-step
// and reused by 4 WMMAs (vmem:wmma ratio 1.25 instead of 2.0).
// A-frag layout (ISA 7.12.2, 32-bit A 16x4): lane<16 -> M=lane, VGPR0=K0,VGPR1=K1;
// lane>=16 -> M=lane-16, VGPR0=K2, VGPR1=K3. B mirrors with N striped over lanes.
// C/D: VGPR r -> M = r + (lane>=16 ? 8 : 0), N = lane&15.
// ---------------------------------------------------------------------------
__global__ __launch_bounds__(64) void mlp_wmma_kernel(
    const float* __restrict__ x,  const float* __restrict__ w1, const float* __restrict__ b1,
    const float* __restrict__ w2, const float* __restrict__ b2, float* __restrict__ z, int n)
{
    __shared__ float h_lds[2][16][HSTR];
    const int wave = threadIdx.x >> 5;
    const int lane = threadIdx.x & 31;
    const int half = lane >> 4;       // 0: lanes 0-15, 1: lanes 16-31
    const int mlan = lane & 15;
    const int row0 = (blockIdx.x * 2 + wave) * 16;
    if (row0 >= n) return;            // wave-uniform; EXEC stays all-ones

    const float* xrow = x + (size_t)(row0 + mlan) * FIN;

    // ---- GEMM1: h[16,256] = relu(x_strip @ w1^T + b1), 4 tiles per pass ----
    for (int ntb = 0; ntb < HID / 64; ++ntb) {
        v8f c0 = {}, c1 = {}, c2 = {}, c3 = {};
        const float* w1r0 = w1 + (size_t)((ntb * 4 + 0) * 16 + mlan) * FIN;
        const float* w1r1 = w1 + (size_t)((ntb * 4 + 1) * 16 + mlan) * FIN;
        const float* w1r2 = w1 + (size_t)((ntb * 4 + 2) * 16 + mlan) * FIN;
        const float* w1r3 = w1 + (size_t)((ntb * 4 + 3) * 16 + mlan) * FIN;
        for (int k = 0; k < FIN; k += 4) {
            const int kk = k + half * 2;
            v2f a, b0, b1v, b2v, b3;
            a.x   = xrow[kk];  a.y   = xrow[kk + 1];
            b0.x  = w1r0[kk];  b0.y  = w1r0[kk + 1];
            b1v.x = w1r1[kk];  b1v.y = w1r1[kk + 1];
            b2v.x = w1r2[kk];  b2v.y = w1r2[kk + 1];
            b3.x  = w1r3[kk];  b3.y  = w1r3[kk + 1];
            c0 = WMMA_F32(a, b0,  c0);
            c1 = WMMA_F32(a, b1v, c1);
            c2 = WMMA_F32(a, b2v, c2);
            c3 = WMMA_F32(a, b3,  c3);
        }
        #pragma unroll
        for (int t = 0; t < 4; ++t) {
            const v8f c = (t == 0) ? c0 : (t == 1) ? c1 : (t == 2) ? c2 : c3;
            const int col = (ntb * 4 + t) * 16 + mlan;
            const float bias = b1[col];
            #pragma unroll
            for (int r = 0; r < 8; ++r)
                h_lds[wave][r + half * 8][col] = fmaxf(c[r] + bias, 0.0f);
        }
    }
    __syncthreads();

    // ---- GEMM2: z[16,64] = h @ w2^T + b2, all 4 tiles in one pass ----
    {
        v8f c0 = {}, c1 = {}, c2 = {}, c3 = {};
        const float* w2r0 = w2 + (size_t)( 0 + mlan) * HID;
        const float* w2r1 = w2 + (size_t)(16 + mlan) * HID;
        const float* w2r2 = w2 + (size_t)(32 + mlan) * HID;
        const float* w2r3 = w2 + (size_t)(48 + mlan) * HID;
        for (int k = 0; k < HID; k += 4) {
            const int kk = k + half * 2;
            v2f a, b0, b1v, b2v, b3;
            a.x   = h_lds[wave][mlan][kk];
            a.y   = h_lds[wave][mlan][kk + 1];
            b0.x  = w2r0[kk];  b0.y  = w2r0[kk + 1];
            b1v.x = w2r1[kk];  b1v.y = w2r1[kk + 1];
            b2v.x = w2r2[kk];  b2v.y = w2r2[kk + 1];
            b3.x  = w2r3[kk];  b3.y  = w2r3[kk + 1];
            c0 = WMMA_F32(a, b0,  c0);
            c1 = WMMA_F32(a, b1v, c1);
            c2 = WMMA_F32(a, b2v, c2);
            c3 = WMMA_F32(a, b3,  c3);
        }
        #pragma unroll
        for (int t = 0; t < 4; ++t) {
            const v8f c = (t == 0) ? c0 : (t == 1) ? c1 : (t == 2) ? c2 : c3;
            const int col = t * 16 + mlan;
            const float bias = b2[col];
            #pragma unroll
            for (int r = 0; r < 8; ++r)
                z[(size_t)(row0 + r + half * 8) * FOUT + col] = c[r] + bias;
        }
    }
}

// ---------------------------------------------------------------------------
// Degree / norm helpers
// ---------------------------------------------------------------------------
__global__ void fill_kernel(float* __restrict__ p, int n, float v) {
    int i = blockIdx.x * blockDim.x + threadIdx.x;
    if (i < n) p[i] = v;
}

__global__ void deg_kernel(const int* __restrict__ cols, float* __restrict__ deg, int E) {
    int i = blockIdx.x * blockDim.x + threadIdx.x;
    int stride = gridDim.x * blockDim.x;
    for (int e = i; e < E; e += stride)
        atomicAdd(&deg[cols[e]], 1.0f);
}

__global__ void rsqrt_kernel(float* __restrict__ dinv, int n) {
    int i = blockIdx.x * blockDim.x + threadIdx.x;
    if (i < n) dinv[i] = rsqrtf(dinv[i]);
}

// ---------------------------------------------------------------------------
// Per-iteration init: y = ALPHA*x0 + BETA*dinv[i]^2*xk  (self-loop folded in).
// Overwrites every element of y (handles poisoned buffers). float4 vectorized.
// ---------------------------------------------------------------------------
__global__ void init_kernel(const float* __restrict__ x0, const float* __restrict__ xs,
                            const float* __restrict__ dinv, float* __restrict__ y, int n) {
    int i = blockIdx.x * blockDim.x + threadIdx.x;     // one float4 of the 16 per node
    if (i >= n * (FOUT / 4)) return;
    int node = i >> 4;
    float d = dinv[node];
    float s = BETA * d * d;
    float4 a = ((const float4*)x0)[i];
    float4 b = ((const float4*)xs)[i];
    float4 o;
    o.x = ALPHA * a.x + s * b.x;
    o.y = ALPHA * a.y + s * b.y;
    o.z = ALPHA * a.z + s * b.z;
    o.w = ALPHA * a.w + s * b.w;
    ((float4*)y)[i] = o;
}

// ---------------------------------------------------------------------------
// Edge scatter: one wave32 per edge; lane handles features {2*lane, 2*lane+1}.
// Gather 256B coalesced from xk[row], scale by BETA*dinv[row]*dinv[col],
// scatter-add into y[col] with non-returning f32 atomics (L2-resident:
// per-iteration working set ~78MB fits the 192MB L2).
// ---------------------------------------------------------------------------
__global__ __launch_bounds__(256) void scatter_kernel(
    const int* __restrict__ rows, const int* __restrict__ cols,
    const float* __restrict__ dinv, const float* __restrict__ xs,
    float* __restrict__ y, int E)
{
    int w      = (blockIdx.x * blockDim.x + threadIdx.x) >> 5;
    int lane   = threadIdx.x & 31;
    int nwaves = (gridDim.x * blockDim.x) >> 5;
    for (int e = w; e < E; e += nwaves) {
        if (lane == 0) {  // stream index arrays ahead into cache
            __builtin_prefetch(rows + e + 8 * nwaves, 0, 1);
            __builtin_prefetch(cols + e + 8 * nwaves, 0, 1);
        }
        int r = rows[e];
        int c = cols[e];
        float s = BETA * dinv[r] * dinv[c];
        float2 xv = *(const float2*)(xs + ((size_t)r << 6) + (lane << 1));
        float* dst = y + ((size_t)c << 6) + (lane << 1);
        atomicAdd(dst,     s * xv.x);
        atomicAdd(dst + 1, s * xv.y);
    }
}

// ---------------------------------------------------------------------------
extern "C" void kernel_launch(void* const* d_in, const int* in_sizes, int n_in,
                              void* d_out, int out_size, void* d_ws, size_t ws_size,
                              hipStream_t stream) {
    const float* x  = (const float*)d_in[0];
    const int*   ei = (const int*)d_in[1];   // int64 in ref, but JAX x64-off -> int32
    const float* w1 = (const float*)d_in[2];
    const float* b1 = (const float*)d_in[3];
    const float* w2 = (const float*)d_in[4];
    const float* b2 = (const float*)d_in[5];
    float* out = (float*)d_out;

    const int n = in_sizes[0] / FIN;   // 100000 (multiple of 32)
    const int E = in_sizes[1] / 2;     // 3200000
    const int* rows = ei;              // edge_index[0] = sources
    const int* cols = ei + E;          // edge_index[1] = targets

    float* x0   = (float*)d_ws;                  // n*64  (z == x0)
    float* buf  = x0  + (size_t)n * FOUT;        // n*64  ping-pong
    float* dinv = buf + (size_t)n * FOUT;        // n     (deg -> rsqrt in place)

    // MLP with f32 WMMA -> x0
    mlp_wmma_kernel<<<(n + 31) / 32, 64, 0, stream>>>(x, w1, b1, w2, b2, x0, n);

    // Degree (self-loop via init=1.0) -> dinv
    fill_kernel<<<(n + 255) / 256, 256, 0, stream>>>(dinv, n, 1.0f);
    deg_kernel<<<4096, 256, 0, stream>>>(cols, dinv, E);
    rsqrt_kernel<<<(n + 255) / 256, 256, 0, stream>>>(dinv, n);

    // K=10 propagation steps; last write lands in d_out (k odd -> out)
    const int elem4 = n * (FOUT / 4);
    for (int k = 0; k < 10; ++k) {
        float*       dst = (k % 2 == 0) ? buf : out;
        const float* src = (k == 0) ? x0 : ((k % 2 == 0) ? out : buf);
        init_kernel<<<(elem4 + 255) / 256, 256, 0, stream>>>(x0, src, dinv, dst, n);
        scatter_kernel<<<40000, 256, 0, stream>>>(rows, cols, dinv, src, dst, E);
    }
}